// GraphSAGELinkPred_79963701117112
// MI455X (gfx1250) — compile-verified
//
#include <hip/hip_runtime.h>

// GraphSAGE link prediction, MI455X (gfx1250, wave32).
// Pipeline (all on `stream`, graph-capture safe):
//   zero(deg+sum) -> scatter(x)+deg -> WMMA-GEMM1(relu) -> zero(sum)
//   -> scatter(h) -> WMMA-GEMM2 -> decode
// GEMMs use V_WMMA_F32_16X16X4_F32 (exact fp32, matches reference numerics).
// Each wave computes a 16x64 output strip (4 WMMA tiles): one A fragment
// feeds 4 matrix ops. LDS weight layout interleaves k-pairs so each B
// fragment is a single aligned ds_load_b64 (no register re-pairing movs).

typedef __attribute__((ext_vector_type(2))) float v2f;
typedef __attribute__((ext_vector_type(8))) float v8f;

#define NNODES 100000
#define CFEAT  128   // feature dim of both SAGE layer inputs
#define NCOLS  64    // output columns per block/wave strip
#define NT     (NCOLS / 16)

// ---------------------------------------------------------------- zero fill
__global__ void zero_kernel(float4* __restrict__ p, long n4) {
  long i = (long)blockIdx.x * blockDim.x + threadIdx.x;
  long s = (long)gridDim.x * blockDim.x;
  for (; i < n4; i += s) p[i] = make_float4(0.f, 0.f, 0.f, 0.f);
}

// ------------------------------------------------- edge scatter (mean prep)
// thread = (edge, feature-quad). float4 gather of src features, 4 fp32
// global atomics into the L2-resident accumulator. Degree fused on q==0.
__global__ void scatter_kernel(const float* __restrict__ feat,
                               const int* __restrict__ ei, long E,
                               float* __restrict__ sum,
                               float* __restrict__ deg, int add_deg) {
  long gid = (long)blockIdx.x * blockDim.x + threadIdx.x;
  if (gid >= E * 32) return;
  long e = gid >> 5;
  int  q = (int)(gid & 31);
  int  s = ei[e];
  int  d = ei[E + e];
  const float4 v = *(const float4*)(feat + (long)s * CFEAT + q * 4);
  float* b = sum + (long)d * CFEAT + q * 4;
  atomicAdd(b + 0, v.x);
  atomicAdd(b + 1, v.y);
  atomicAdd(b + 2, v.z);
  atomicAdd(b + 3, v.w);
  if (add_deg && q == 0) atomicAdd(deg + d, 1.0f);
}

// --------------------------------------------- fused SAGE layer, WMMA fp32
// out[m,n] = relu?( sum_k mean[m,k]*Wl[n,k] + sum_k xin[m,k]*Wr[n,k] + b[n] )
// mean[m,k] = sum[m,k] / max(deg[m],1) computed on the fly.
// Block: 256 thr = 8 waves -> 128 rows x 64 cols; wave -> 16 rows x 64 cols.
// LDS layout: element (k,n) at ((k>>1)*NCOLS + n)*2 + (k&1) so the B
// fragment {W[k],W[k+1]} for any (k-step, col-tile) is one aligned b64.
__global__ __launch_bounds__(256)
void sage_gemm_wmma(const float* __restrict__ sum, const float* __restrict__ deg,
                    const float* __restrict__ xin,
                    const float* __restrict__ Wl, const float* __restrict__ Wr,
                    const float* __restrict__ bias,
                    float* __restrict__ out, int M, int NOUT, int do_relu) {
  __shared__ float wlds[256 * NCOLS];         // 64 KB, k-pair interleaved

  const int tid = threadIdx.x;
  const int n0  = blockIdx.y * NCOLS;

  // Stage the NCOLS-column slice of the combined [Wl ; Wr] weight (K=256).
  for (int i = tid; i < 256 * NCOLS; i += 256) {
    int k = i >> 6, n = i & (NCOLS - 1);
    float w = (k < 128) ? Wl[(n0 + n) * 128 + k]
                        : Wr[(n0 + n) * 128 + (k - 128)];
    wlds[(((k >> 1) * NCOLS) + n) * 2 + (k & 1)] = w;
  }
  __syncthreads();

  const int wave  = tid >> 5;
  const int lane  = tid & 31;
  const int laneM = lane & 15;                // A row / B col / C col
  const int kh    = lane >> 4;                // K half-select (ISA layout)
  const int row0  = blockIdx.x * 128 + wave * 16;

  int rowA = row0 + laneM;
  int rowC = rowA < M ? rowA : (M - 1);       // clamp loads, keep EXEC full
  const float invdeg = 1.0f / fmaxf(deg[rowC], 1.0f);
  const float* sP = sum + (long)rowC * CFEAT;
  const float* xP = xin + (long)rowC * CFEAT;

  // C/D layout: lanes0-15 -> M=r, lanes16-31 -> M=8+r; N = lane&15.
  v8f acc[NT];
#pragma unroll
  for (int nt = 0; nt < NT; ++nt) {
    const float bcol = bias[n0 + nt * 16 + laneM];
#pragma unroll
    for (int r = 0; r < 8; ++r) acc[nt][r] = bcol;
  }

  const int ko = 2 * kh;                      // per-lane K offset in each k4
  // per-lane base (in k-pair rows) for B fragments: row = (k0+ko)>>1
  const v2f* wp = (const v2f*)&wlds[0];

  // K = 0..127 : mean half (scaled neighbor sums)
#pragma unroll 4
  for (int k0 = 0; k0 < 128; k0 += 4) {
    v2f a = *(const v2f*)(sP + k0 + ko);
    a = a * invdeg;
    const int kr = (k0 >> 1) + kh;            // k-pair row index
#pragma unroll
    for (int nt = 0; nt < NT; ++nt) {
      v2f b = wp[kr * NCOLS + nt * 16 + laneM];
      acc[nt] = __builtin_amdgcn_wmma_f32_16x16x4_f32(
          false, a, false, b, (short)0, acc[nt], false, false);
    }
  }
  // K = 128..255 : root half (self features)
#pragma unroll 4
  for (int k0 = 0; k0 < 128; k0 += 4) {
    v2f a = *(const v2f*)(xP + k0 + ko);
    const int kr = ((128 + k0) >> 1) + kh;
#pragma unroll
    for (int nt = 0; nt < NT; ++nt) {
      v2f b = wp[kr * NCOLS + nt * 16 + laneM];
      acc[nt] = __builtin_amdgcn_wmma_f32_16x16x4_f32(
          false, a, false, b, (short)0, acc[nt], false, false);
    }
  }

#pragma unroll
  for (int nt = 0; nt < NT; ++nt) {
#pragma unroll
    for (int r = 0; r < 8; ++r) {
      int mrow = row0 + kh * 8 + r;
      if (mrow < M) {
        float v = acc[nt][r];
        if (do_relu) v = fmaxf(v, 0.f);
        out[(long)mrow * NOUT + n0 + nt * 16 + laneM] = v;
      }
    }
  }
}

// ------------------------------------------------------------------ decode
// out[p] = z[s]·Wdec[0:64] + z[d]·Wdec[64:128] + bdec
__global__ void decode_kernel(const float* __restrict__ z,
                              const int* __restrict__ eli, long P,
                              const float* __restrict__ Wdec,
                              const float* __restrict__ bdec,
                              float* __restrict__ out) {
  long p = (long)blockIdx.x * blockDim.x + threadIdx.x;
  if (p >= P) return;
  int s = eli[p];
  int d = eli[P + p];
  const float4* zs = (const float4*)(z + (long)s * 64);
  const float4* zd = (const float4*)(z + (long)d * 64);
  const float4* w0 = (const float4*)(Wdec);
  const float4* w1 = (const float4*)(Wdec + 64);
  float acc = bdec[0];
#pragma unroll
  for (int i = 0; i < 16; ++i) {
    float4 a = zs[i], wa = w0[i];
    acc += a.x * wa.x + a.y * wa.y + a.z * wa.z + a.w * wa.w;
    float4 b = zd[i], wb = w1[i];
    acc += b.x * wb.x + b.y * wb.y + b.z * wb.z + b.w * wb.w;
  }
  out[p] = acc;
}

// ------------------------------------------------------------------ driver
extern "C" void kernel_launch(void* const* d_in, const int* in_sizes, int n_in,
                              void* d_out, int out_size, void* d_ws, size_t ws_size,
                              hipStream_t stream) {
  const float* x    = (const float*)d_in[0];
  const int*   ei   = (const int*)  d_in[1];
  const int*   eli  = (const int*)  d_in[2];
  const float* Wl1  = (const float*)d_in[3];
  const float* bl1  = (const float*)d_in[4];
  const float* Wr1  = (const float*)d_in[5];
  const float* Wl2  = (const float*)d_in[6];
  const float* bl2  = (const float*)d_in[7];
  const float* Wr2  = (const float*)d_in[8];
  const float* Wdec = (const float*)d_in[9];
  const float* bdec = (const float*)d_in[10];

  const long E = in_sizes[1] / 2;
  const long P = in_sizes[2] / 2;
  const int  M = NNODES;

  // workspace carve-up (floats): deg | sum | h | z
  const long DEGF = 102400;                 // 100000 rounded up (alignment)
  const long SUMF = (long)M * CFEAT;        // 12.8M
  char* ws = (char*)d_ws;
  float* deg  = (float*)(ws);
  float* sum  = (float*)(ws + DEGF * 4);
  float* hbuf = (float*)(ws + (DEGF + SUMF) * 4);
  float* zbuf = (float*)(ws + (DEGF + 2 * SUMF) * 4);

  const int ZB = 4096;

  // ---- layer 1 ----
  zero_kernel<<<ZB, 256, 0, stream>>>((float4*)deg, (DEGF + SUMF) / 4);
  {
    long total = E * 32;
    int blocks = (int)((total + 255) / 256);
    scatter_kernel<<<blocks, 256, 0, stream>>>(x, ei, E, sum, deg, 1);
  }
  {
    dim3 grid((M + 127) / 128, 128 / NCOLS);
    sage_gemm_wmma<<<grid, 256, 0, stream>>>(sum, deg, x, Wl1, Wr1, bl1,
                                             hbuf, M, 128, 1);
  }

  // ---- layer 2 ----
  zero_kernel<<<ZB, 256, 0, stream>>>((float4*)sum, SUMF / 4);
  {
    long total = E * 32;
    int blocks = (int)((total + 255) / 256);
    scatter_kernel<<<blocks, 256, 0, stream>>>(hbuf, ei, E, sum, deg, 0);
  }
  {
    dim3 grid((M + 127) / 128, 64 / NCOLS);
    sage_gemm_wmma<<<grid, 256, 0, stream>>>(sum, deg, hbuf, Wl2, Wr2, bl2,
                                             zbuf, M, 64, 0);
  }

  // ---- decode ----
  {
    int blocks = (int)((P + 255) / 256);
    decode_kernel<<<blocks, 256, 0, stream>>>(zbuf, eli, P, Wdec, bdec,
                                              (float*)d_out);
  }
}